// DensityLoss_20409684590745
// MI455X (gfx1250) — compile-verified
//
#include <hip/hip_runtime.h>
#include <math.h>

typedef __attribute__((ext_vector_type(2))) float v2f;
typedef __attribute__((ext_vector_type(4))) float v4f;
typedef __attribute__((ext_vector_type(8))) float v8f;

#define NPTS 8192
#define NB   4
#define KNN  10
#define SENT 3.0e30f
#define TPAD 20   // tile row stride: 16B-aligned rows, conflict-free mod 64 banks

// One wave (32 lanes) per 16-row tile of one batch.
// Per row i, d^2 = |i|^2 + |j|^2 - 2<i,j> is monotonic in h = |j|^2 - 2<i,j>,
// so top-k selection runs on h; sqrt(|i|^2 + h) is applied only to the 10
// finalists at the end.
__global__ __launch_bounds__(32)
void knn_avg_kernel(const float* __restrict__ pc, float* __restrict__ avg_out) {
    const int it   = blockIdx.x;        // i-tile, 0..511
    const int b    = blockIdx.y;        // batch
    const int lane = threadIdx.x;       // 0..31
    const int hi   = lane >> 4;         // half-wave select
    const int p    = lane & 15;

    const float* base = pc + (size_t)b * NPTS * 3;
    const int i0 = it * 16;

    __shared__ __align__(16) float tile[2][16][TPAD];
    __shared__ float lists[32][KNN];

    // ---- A operand (16x4 f32, ISA 7.12.2 layout):
    // lanes 0-15: VGPR0=K0(x), VGPR1=K1(y) of row M=lane
    // lanes16-31: VGPR0=K2(z), VGPR1=K3(0) of row M=lane-16
    const int ia = i0 + p;
    const float aix = base[ia*3+0], aiy = base[ia*3+1], aiz = base[ia*3+2];
    v2f A;
    A.x = hi ? aiz : aix;
    A.y = hi ? 0.0f : aiy;

    // |i|^2 of the row this lane owns in the SELECTION phase (row i0+p):
    // same point we just loaded for the A operand.
    const float sqrow = aix*aix + aiy*aiy + aiz*aiz;

    // ---- sorted ascending top-K list of h = |j|^2 - 2<i,j> per lane
    float topv[KNN];
#pragma unroll
    for (int t = 0; t < KNN; ++t) topv[t] = SENT;

    // Process one pair of j-tiles. diag0/diag1 are literal constants at the
    // hot call sites, so the self-exclusion code constant-folds away there.
    auto body = [&](int jt, bool diag0, bool diag1) __attribute__((always_inline)) {
        const int jc0 = jt*16 + p;
        const int jc1 = jc0 + 16;

        const float b0x = base[jc0*3+0], b0y = base[jc0*3+1], b0z = base[jc0*3+2];
        const float b1x = base[jc1*3+0], b1y = base[jc1*3+1], b1z = base[jc1*3+2];
        const float sqj0 = b0x*b0x + b0y*b0y + b0z*b0z;
        const float sqj1 = b1x*b1x + b1y*b1y + b1z*b1z;

        v2f B0, B1;
        B0.x = hi ? b0z : b0x;  B0.y = hi ? 0.0f : b0y;
        B1.x = hi ? b1z : b1x;  B1.y = hi ? 0.0f : b1y;

        v8f acc0 = {};
        v8f acc1 = {};
        acc0 = __builtin_amdgcn_wmma_f32_16x16x4_f32(
            false, A, false, B0, (short)0, acc0, false, false);
        acc1 = __builtin_amdgcn_wmma_f32_16x16x4_f32(
            false, A, false, B1, (short)0, acc1, false, false);

        // h = |j|^2 - 2<i,j>: one v_fma per element, staged through LDS
#pragma unroll
        for (int r = 0; r < 8; ++r) {
            const int m = r + 8*hi;     // row within tile
            tile[0][m][p] = fmaf(-2.0f, acc0[r], sqj0);
            tile[1][m][p] = fmaf(-2.0f, acc1[r], sqj1);
        }
        if (diag0) { if (lane < 16) tile[0][p][p] = SENT; }  // exclude self
        if (diag1) { if (lane < 16) tile[1][p][p] = SENT; }
        __syncthreads();

        // selection: lane owns row p, columns [hi*8, hi*8+8) of both sub-tiles.
        // Two b128 loads per sub-tile half, then guards run from registers.
#pragma unroll
        for (int s = 0; s < 2; ++s) {
            const v4f* rowv = (const v4f*)(&tile[s][p][hi * 8]);
            const v4f c0 = rowv[0];
            const v4f c1 = rowv[1];
            float cand[8] = {c0.x, c0.y, c0.z, c0.w, c1.x, c1.y, c1.z, c1.w};
#pragma unroll
            for (int t = 0; t < 8; ++t) {
                const float c = cand[t];
                // wave-wide early out: almost every candidate is worse than
                // every lane's current 10th-smallest in steady state.
                if (__any(c < topv[KNN-1])) {
                    topv[KNN-1] = fminf(topv[KNN-1], c);
#pragma unroll
                    for (int u = KNN-1; u >= 1; --u) {
                        const float lo = fminf(topv[u-1], topv[u]);
                        const float hh = fmaxf(topv[u-1], topv[u]);
                        topv[u-1] = lo;
                        topv[u]   = hh;
                    }
                }
            }
        }
        __syncthreads();
    };

    // ---- three phases: the diagonal pair is handled exactly once, so the
    // hot loops carry zero self-exclusion code.
    const int dpair = it & ~1;
    for (int jt = 0; jt < dpair; jt += 2)
        body(jt, false, false);
    body(dpair, (it & 1) == 0, (it & 1) == 1);
    for (int jt = dpair + 2; jt < NPTS/16; jt += 2)
        body(jt, false, false);

    // ---- convert finalists to distances: sqrt(|i|^2 + h), monotonic so the
    // sorted order is preserved; then merge lane L (cols 0-7) with lane L+16
    // (cols 8-15) per row.
#pragma unroll
    for (int t = 0; t < KNN; ++t)
        lists[lane][t] = __builtin_amdgcn_sqrtf(fmaxf(sqrow + topv[t], 0.0f));
    __syncthreads();

    if (lane < 16) {
        const float* la = lists[lane];
        const float* lb = lists[lane + 16];
        int x = 0, y = 0;
        float s = 0.0f;
#pragma unroll
        for (int t = 0; t < KNN; ++t) {   // 10 picks from two sorted 10-lists
            const float va = la[x], vb = lb[y];
            if (va <= vb) { s += va; ++x; } else { s += vb; ++y; }
        }
        avg_out[b * NPTS + i0 + lane] = s * (1.0f / KNN);
    }
}

// Per-batch unbiased variance of avg_out[b][*]
__global__ __launch_bounds__(256)
void var_kernel(const float* __restrict__ avg, float* __restrict__ var_out) {
    const int b = blockIdx.x;
    const int tid = threadIdx.x;
    __shared__ float s1s[256];
    __shared__ float s2s[256];
    float s1 = 0.0f, s2 = 0.0f;
    for (int i = tid; i < NPTS; i += 256) {
        const float v = avg[b * NPTS + i];
        s1 += v;
        s2 += v * v;
    }
    s1s[tid] = s1; s2s[tid] = s2;
    __syncthreads();
    for (int off = 128; off > 0; off >>= 1) {
        if (tid < off) { s1s[tid] += s1s[tid+off]; s2s[tid] += s2s[tid+off]; }
        __syncthreads();
    }
    if (tid == 0) {
        const float S1 = s1s[0], S2 = s2s[0];
        var_out[b] = (S2 - S1 * S1 / (float)NPTS) / (float)(NPTS - 1);
    }
}

__global__ void final_kernel(const float* __restrict__ v, float* __restrict__ out) {
    if (threadIdx.x == 0 && blockIdx.x == 0)
        out[0] = 0.25f * (v[0] + v[1] + v[2] + v[3]);
}

extern "C" void kernel_launch(void* const* d_in, const int* in_sizes, int n_in,
                              void* d_out, int out_size, void* d_ws, size_t ws_size,
                              hipStream_t stream) {
    const float* pc = (const float*)d_in[0];   // [4, 8192, 3] f32
    float* ws   = (float*)d_ws;
    float* avgs = ws;                 // 32768 floats
    float* vars = ws + NB * NPTS;     // 4 floats

    dim3 g1(NPTS / 16, NB);
    knn_avg_kernel<<<g1, 32, 0, stream>>>(pc, avgs);
    var_kernel<<<NB, 256, 0, stream>>>(avgs, vars);
    final_kernel<<<1, 32, 0, stream>>>(vars, (float*)d_out);
}